// SSMScan1D_52845277610275
// MI455X (gfx1250) — compile-verified
//
#include <hip/hip_runtime.h>
#include <cstdint>
#include <cstddef>

// Problem constants (from the reference)
#define B_    8
#define L_    4096
#define C_    1024
#define TC    256              // timesteps per chunk
#define NCH   (L_ / TC)        // 16 chunks
#define SLAB  128              // channels per block in K3 (8 waves x 16)
#define NSLAB (C_ / SLAB)      // 8
#define RT    32               // rows per TDM transfer (2 WMMA subtiles)
#define NBUF  (TC / RT)        // 8 transfers per chunk

typedef __attribute__((ext_vector_type(2))) float        v2f;
typedef __attribute__((ext_vector_type(8))) float        v8f;
typedef __attribute__((ext_vector_type(4))) unsigned int v4u;
typedef __attribute__((ext_vector_type(4))) int          v4i;
typedef __attribute__((ext_vector_type(8))) int          v8i;

#if defined(__gfx1250__) && __has_builtin(__builtin_amdgcn_tensor_load_to_lds) && \
    __has_builtin(__builtin_amdgcn_s_wait_tensorcnt)
#define HAVE_TDM 1
#else
#define HAVE_TDM 0
#endif

#if defined(__gfx1250__) && __has_builtin(__builtin_amdgcn_wmma_f32_16x16x4_f32)
#define HAVE_WMMA 1
#else
#define HAVE_WMMA 0
#endif

#if HAVE_TDM
// Issue one TDM load: 2D tile of 16 channels x RT rows of f32, row stride C_.
// D# per CDNA5 ISA ch.8: group0 = {flags, lds_addr, gaddr_lo, gaddr_hi|type=2},
// group1 = {data_size, tensor dims/strides, tile dims}. Groups 2/3 unused (2D).
// This toolchain exposes the 6-arg builtin form: (v4u, v8i, v4i, v4i, v8i, cpol).
__device__ __forceinline__ void tdm_issue(unsigned lds_addr, unsigned long long gaddr) {
  v4u g0;
  g0.x = 1u;                                                            // count=1, user descriptor
  g0.y = (unsigned)__builtin_amdgcn_readfirstlane((int)lds_addr);       // lds_addr (bytes)
  g0.z = (unsigned)__builtin_amdgcn_readfirstlane((int)(unsigned)(gaddr & 0xffffffffull));
  g0.w = (unsigned)__builtin_amdgcn_readfirstlane(
             (int)((unsigned)((gaddr >> 32) & 0x01ffffffull) | (2u << 30))); // type=2 ("image")
  v8i g1;
  g1[0] = (int)(2u << 16);            // data_size = 2 -> 4 bytes; no multicast/iterate/pad
  g1[1] = (int)(16u << 16);           // tensor_dim0 = 16 (channels)
  g1[2] = (int)((unsigned)RT << 16);  // tensor_dim1 = RT rows
  g1[3] = (int)(16u << 16);           // tile_dim0 = 16
  g1[4] = (int)RT;                    // tile_dim1 = RT, tile_dim2 = 0
  g1[5] = (int)C_;                    // tensor_dim0_stride = C elements
  g1[6] = 0;
  g1[7] = 0;
  v4i z4 = {0, 0, 0, 0};
  v8i z8 = {0, 0, 0, 0, 0, 0, 0, 0};
  __builtin_amdgcn_tensor_load_to_lds(g0, g1, z4, z4, z8, 0);
}
#endif

// ---------------- K1: chunk-end states (one streaming read of x) ----------------
__global__ __launch_bounds__(256) void ssm_k1(const float* __restrict__ x,
                                              const float* __restrict__ alpha,
                                              const float* __restrict__ beta,
                                              float* __restrict__ send) {
  const int chunk = blockIdx.x;
  const int b     = blockIdx.y;
  const int c4    = (int)threadIdx.x;            // channels [4*c4, 4*c4+4)
  const float4 al = ((const float4*)alpha)[c4];
  const float4 bt = ((const float4*)beta)[c4];
  float4 a;
  a.x = 1.0f / (1.0f + __expf(-al.x));
  a.y = 1.0f / (1.0f + __expf(-al.y));
  a.z = 1.0f / (1.0f + __expf(-al.z));
  a.w = 1.0f / (1.0f + __expf(-al.w));

  const float4* xp = (const float4*)(x + ((size_t)b * L_ + (size_t)chunk * TC) * C_) + c4;
  float4 s = {0.f, 0.f, 0.f, 0.f};
  constexpr int PF = 8;                           // register prefetch depth
  float4 pf[PF];
#pragma unroll
  for (int i = 0; i < PF; ++i) pf[i] = xp[(size_t)i * (C_ / 4)];
#pragma unroll 8
  for (int t = 0; t < TC; ++t) {
    const float4 v = pf[t & (PF - 1)];
    if (t + PF < TC) pf[t & (PF - 1)] = xp[(size_t)(t + PF) * (C_ / 4)];
    s.x = fmaf(a.x, s.x, bt.x * v.x);
    s.y = fmaf(a.y, s.y, bt.y * v.y);
    s.z = fmaf(a.z, s.z, bt.z * v.z);
    s.w = fmaf(a.w, s.w, bt.w * v.w);
  }
  ((float4*)send)[((size_t)b * NCH + chunk) * (C_ / 4) + c4] = s;
}

// ---------------- K2: scan of the NCH chunk carries (tiny) ----------------
__global__ __launch_bounds__(256) void ssm_k2(const float* __restrict__ alpha,
                                              const float* __restrict__ send,
                                              float* __restrict__ carry) {
  const int idx = (int)(blockIdx.x * 256 + threadIdx.x);   // [0, B_*C_)
  const int b = idx / C_, c = idx % C_;
  const float a = 1.0f / (1.0f + __expf(-alpha[c]));
  float aT = a;
#pragma unroll
  for (int i = 0; i < 8; ++i) aT *= aT;                    // a^256 == a^TC
  float s = 0.0f;
  for (int j = 0; j < NCH; ++j) {
    const size_t o = ((size_t)b * NCH + j) * C_ + c;
    carry[o] = s;                                          // state entering chunk j
    s = fmaf(aT, s, send[o]);
  }
}

// ---------------- K3: TDM-fed, WMMA-scan output kernel ----------------
__global__ __launch_bounds__(256) void ssm_k3(const float* __restrict__ x,
                                              const float* __restrict__ alpha,
                                              const float* __restrict__ beta,
                                              const float* __restrict__ gamma_,
                                              const float* __restrict__ delta,
                                              const float* __restrict__ carry,
                                              float* __restrict__ y) {
  __shared__ float smem[2][8][RT][16];                     // 32 KB, per-wave double buffers
  const int chunk = blockIdx.x;
  const int slab  = blockIdx.y;
  const int b     = blockIdx.z;
  const int lane  = (int)(threadIdx.x & 31u);
  const int wave  = (int)(threadIdx.x >> 5);
  const int n = lane & 15;                                 // channel within wave's 16-slab
  const int h = lane >> 4;                                 // wave half (splits K / M)
  const int c0 = slab * SLAB + wave * 16;
  const int c  = c0 + n;
  const int t0 = chunk * TC;

  const float bet = beta[c];
  const float gam = gamma_[c];
  const float del = delta[c];
  const float r   = 1.0f + __expf(-alpha[c]);              // exactly 1/a
  const float a   = 1.0f / r;
  const float a2 = a * a, a4 = a2 * a2, a8 = a4 * a4, a16 = a8 * a8;
  const float r2 = r * r, r4 = r2 * r2, r8 = r4 * r4;

  // z_k = beta * a^{-k} * x_k ; this lane's B-layout k = 4j + 2h + v
  const float zb = bet * (h ? r2 : 1.0f);
  float zs[8] = {zb,           zb * r,
                 zb * r4,      zb * r4 * r,
                 zb * r8,      zb * r8 * r,
                 zb * r8 * r4, zb * r8 * r4 * r};
  // a^t for this lane's D-layout t = rI + 8h
  const float ab = h ? a8 : 1.0f;
  float ap[8] = {ab,           ab * a,
                 ab * a2,      ab * a2 * a,
                 ab * a4,      ab * a4 * a,
                 ab * a4 * a2, ab * a4 * a2 * a};

#if HAVE_WMMA
  // Lower-triangular all-ones cumsum matrix, A-operand layout (lanes=M rows,
  // halves split K): element (m, k=4j+2h+v) = (k <= m).
  v2f A0, A1, A2, A3;
  A0.x = (0 + 2 * h <= n) ? 1.0f : 0.0f;  A0.y = (1 + 2 * h <= n) ? 1.0f : 0.0f;
  A1.x = (4 + 2 * h <= n) ? 1.0f : 0.0f;  A1.y = (5 + 2 * h <= n) ? 1.0f : 0.0f;
  A2.x = (8 + 2 * h <= n) ? 1.0f : 0.0f;  A2.y = (9 + 2 * h <= n) ? 1.0f : 0.0f;
  A3.x = (12 + 2 * h <= n) ? 1.0f : 0.0f; A3.y = (13 + 2 * h <= n) ? 1.0f : 0.0f;
#endif

  float s_in = carry[((size_t)b * NCH + chunk) * C_ + c];

  // Consume one staged RT x 16 buffer = two 16(time) x 16(channel) WMMA tiles.
  auto consume = [&](const float* __restrict__ buf, int trow) {
#pragma unroll
    for (int sub = 0; sub < 2; ++sub) {
      const int row0 = sub * 16;
      float sloc[8];                                       // local s_t (zero carry-in)
#if HAVE_WMMA
      v8f acc = {};
      v2f Bv;
      Bv.x = buf[(row0 + 0 + 2 * h) * 16 + n] * zs[0];
      Bv.y = buf[(row0 + 1 + 2 * h) * 16 + n] * zs[1];
      acc = __builtin_amdgcn_wmma_f32_16x16x4_f32(false, A0, false, Bv, (short)0, acc, false, false);
      Bv.x = buf[(row0 + 4 + 2 * h) * 16 + n] * zs[2];
      Bv.y = buf[(row0 + 5 + 2 * h) * 16 + n] * zs[3];
      acc = __builtin_amdgcn_wmma_f32_16x16x4_f32(false, A1, false, Bv, (short)0, acc, false, false);
      Bv.x = buf[(row0 + 8 + 2 * h) * 16 + n] * zs[4];
      Bv.y = buf[(row0 + 9 + 2 * h) * 16 + n] * zs[5];
      acc = __builtin_amdgcn_wmma_f32_16x16x4_f32(false, A2, false, Bv, (short)0, acc, false, false);
      Bv.x = buf[(row0 + 12 + 2 * h) * 16 + n] * zs[6];
      Bv.y = buf[(row0 + 13 + 2 * h) * 16 + n] * zs[7];
      acc = __builtin_amdgcn_wmma_f32_16x16x4_f32(false, A3, false, Bv, (short)0, acc, false, false);
#pragma unroll
      for (int rI = 0; rI < 8; ++rI) sloc[rI] = ap[rI] * acc[rI];
#else
      // VALU fallback: direct exponentially-weighted prefix per lane.
#pragma unroll
      for (int rI = 0; rI < 8; ++rI) {
        const int t = rI + 8 * h;
        float w = bet, s = 0.0f;
        for (int k = t; k >= 0; --k) { s = fmaf(w, buf[(row0 + k) * 16 + n], s); w *= a; }
        sloc[rI] = s;
      }
#endif
      float* yp = y + (((size_t)b * L_ + (size_t)(trow + row0 + 8 * h)) * C_ + c);
#pragma unroll
      for (int rI = 0; rI < 8; ++rI) {
        const float xv = buf[(row0 + rI + 8 * h) * 16 + n];
        const float s  = sloc[rI] + ap[rI] * a * s_in;     // + a^{t+1} * carry
        yp[(size_t)rI * C_] = gam * s + del * xv;
      }
      // Advance carry by 16 steps: s' = a^16 * s_in + s_local(t=15).
      // s_local(15) lives on upper-half lanes (rI=7, h=1): broadcast lane (16+n).
      const float tail = sloc[7];
      const float tb   = __int_as_float(
          __builtin_amdgcn_ds_bpermute((16 + n) << 2, __float_as_int(tail)));
      s_in = fmaf(a16, s_in, tb);
    }
  };

#if HAVE_TDM
  const unsigned lds_base =
      (unsigned)(uint64_t)(uintptr_t)&smem[0][0][0][0];
  const unsigned wave_lds0 = lds_base + (unsigned)((0 * 8 + wave) * RT * 16 * 4);
  const unsigned wave_lds1 = lds_base + (unsigned)((1 * 8 + wave) * RT * 16 * 4);
  const unsigned long long xbase =
      (unsigned long long)(uintptr_t)x +
      4ull * (((unsigned long long)b * L_ + (unsigned long long)t0) * C_ + (unsigned long long)c0);

  tdm_issue(wave_lds0, xbase + 0ull * RT * C_ * 4ull);     // prologue: fill both buffers
  tdm_issue(wave_lds1, xbase + 1ull * RT * C_ * 4ull);
  for (int it = 0; it < NBUF - 1; ++it) {
    __builtin_amdgcn_s_wait_tensorcnt(1);                  // buffer `it` landed in LDS
    consume(&smem[it & 1][wave][0][0], t0 + it * RT);
    if (it + 2 < NBUF)
      tdm_issue((it & 1) ? wave_lds1 : wave_lds0,
                xbase + (unsigned long long)(it + 2) * RT * C_ * 4ull);
  }
  __builtin_amdgcn_s_wait_tensorcnt(0);
  consume(&smem[(NBUF - 1) & 1][wave][0][0], t0 + (NBUF - 1) * RT);
#else
  for (int it = 0; it < NBUF; ++it) {
    float* buf = &smem[it & 1][wave][0][0];
    for (int idx = lane; idx < RT * 16; idx += 32) {
      const int rr = idx >> 4, cc = idx & 15;
      buf[idx] = x[((size_t)b * L_ + (size_t)(t0 + it * RT + rr)) * C_ + c0 + cc];
    }
    consume(buf, t0 + it * RT);
  }
#endif
}

extern "C" void kernel_launch(void* const* d_in, const int* in_sizes, int n_in,
                              void* d_out, int out_size, void* d_ws, size_t ws_size,
                              hipStream_t stream) {
  (void)in_sizes; (void)n_in; (void)out_size; (void)ws_size;
  const float* x     = (const float*)d_in[0];
  const float* alpha = (const float*)d_in[1];
  const float* beta  = (const float*)d_in[2];
  const float* gam   = (const float*)d_in[3];
  const float* del   = (const float*)d_in[4];
  float* y = (float*)d_out;

  float* send  = (float*)d_ws;                      // [B, NCH, C] chunk-end states
  float* carry = send + (size_t)B_ * NCH * C_;      // [B, NCH, C] chunk carry-in

  ssm_k1<<<dim3(NCH, B_), 256, 0, stream>>>(x, alpha, beta, send);
  ssm_k2<<<(B_ * C_) / 256, 256, 0, stream>>>(alpha, send, carry);
  ssm_k3<<<dim3(NCH, NSLAB, B_), 256, 0, stream>>>(x, alpha, beta, gam, del, carry, y);
}